// GCNEncoder_68092411511097
// MI455X (gfx1250) — compile-verified
//
#include <hip/hip_runtime.h>

typedef __attribute__((ext_vector_type(2))) float v2f;
typedef __attribute__((ext_vector_type(8))) float v8f;

// ---------------------------------------------------------------------------
// Degree / normalization kernels
// ---------------------------------------------------------------------------
__global__ void deg_init_kernel(float* __restrict__ deg, int n) {
    int i = blockIdx.x * blockDim.x + threadIdx.x;
    if (i < n) deg[i] = 1.0f;  // self-loop contributes 1 to every node's degree
}

__global__ void deg_count_kernel(const int* __restrict__ dst, float* __restrict__ deg, int nE) {
    int e = blockIdx.x * blockDim.x + threadIdx.x;
    if (e < nE) atomicAdd(&deg[dst[e]], 1.0f);
}

__global__ void deg_to_dinv_kernel(float* __restrict__ deg, int n) {
    int i = blockIdx.x * blockDim.x + threadIdx.x;
    if (i < n) deg[i] = rsqrtf(deg[i]);   // deg >= 1 always (self-loops)
}

// ---------------------------------------------------------------------------
// GEMM1: C[N,32] = A[N,32] @ W[32,32] using V_WMMA_F32_16X16X4_F32.
// One wave computes a 16-row x 32-col tile (two 16x16 accumulators sharing A).
// N is a multiple of 16 (100000 = 6250*16), so no tail handling needed.
// ---------------------------------------------------------------------------
__global__ void gemm_xw1_kernel(const float* __restrict__ A, const float* __restrict__ W,
                                float* __restrict__ C, int nRows) {
    int waveId = (blockIdx.x * blockDim.x + threadIdx.x) >> 5;
    int lane   = threadIdx.x & 31;
    int m0     = waveId * 16;
    if (m0 >= nRows) return;               // wave-uniform: EXEC stays all-1s for WMMA

    int row  = m0 + (lane & 15);
    int grp  = lane >> 4;                  // 0 or 1
    int koff = grp * 2;
    int col  = lane & 15;

    v8f acc0 = {};
    v8f acc1 = {};
#pragma unroll
    for (int k = 0; k < 8; ++k) {          // K = 32 = 8 * 4
        int kb = 4 * k + koff;
        // A 16x4 fragment: lane holds A[row][kb], A[row][kb+1] (8B-aligned)
        v2f a = *reinterpret_cast<const v2f*>(A + row * 32 + kb);
        // B 4x16 fragments: VGPR j holds B[K=kb+j][col]
        v2f b0, b1;
        b0.x = W[kb * 32 + col];        b0.y = W[(kb + 1) * 32 + col];
        b1.x = W[kb * 32 + 16 + col];   b1.y = W[(kb + 1) * 32 + 16 + col];
        acc0 = __builtin_amdgcn_wmma_f32_16x16x4_f32(false, a, false, b0, (short)0, acc0, false, false);
        acc1 = __builtin_amdgcn_wmma_f32_16x16x4_f32(false, a, false, b1, (short)0, acc1, false, false);
    }

    int rowbase = m0 + grp * 8;            // C layout: VGPR r -> M = r (+8 for hi lanes)
#pragma unroll
    for (int r = 0; r < 8; ++r) {
        C[(rowbase + r) * 32 + col]      = acc0[r];
        C[(rowbase + r) * 32 + 16 + col] = acc1[r];
    }
}

// ---------------------------------------------------------------------------
// Self-loop initialization: agg[i][c] = feat[i][c] * dinv[i]^2
// ---------------------------------------------------------------------------
__global__ void self_init_kernel(const float* __restrict__ feat, const float* __restrict__ dinv,
                                 float* __restrict__ agg, int n32) {
    int idx = blockIdx.x * blockDim.x + threadIdx.x;
    if (idx < n32) {
        float di = dinv[idx >> 5];
        agg[idx] = feat[idx] * di * di;
    }
}

// ---------------------------------------------------------------------------
// Edge aggregation: one wave processes a batch of 32 edges; lane = channel.
// Phase 1 (coalesced): lane e loads src/dst/norm for edge base+lane.
// Phase 2: iterate the 32 edges, broadcasting s/d/norm via v_readlane (SGPR
// lane select), so the feature gather and atomic scatter use a uniform SGPR
// base + lane offset: exactly one 128B line in, one 128B line of f32 atomics
// out, per edge. All of it is L2-resident on MI455X (12.8MB features +
// 12.8MB edge index << 192MB L2), so VMEM issue rate is the bound and we
// cut per-edge VMEM lane-ops from 6 to 2.
// ---------------------------------------------------------------------------
__global__ void scatter_agg_kernel(const int* __restrict__ src, const int* __restrict__ dst,
                                   const float* __restrict__ dinv, const float* __restrict__ feat,
                                   float* __restrict__ agg, int nEdges) {
    int lane = threadIdx.x & 31;
    int wave = (blockIdx.x * blockDim.x + threadIdx.x) >> 5;
    int base = wave * 32;
    if (base >= nEdges) return;

    int  myE   = base + lane;
    bool valid = myE < nEdges;
    int   sL = valid ? src[myE] : 0;
    int   dL = valid ? dst[myE] : 0;
    float wL = valid ? dinv[sL] * dinv[dL] : 0.0f;

    int cnt = nEdges - base;
    if (cnt > 32) cnt = 32;

#pragma unroll 4
    for (int i = 0; i < cnt; ++i) {
        int   s = __builtin_amdgcn_readlane(sL, i);
        int   d = __builtin_amdgcn_readlane(dL, i);
        float w = __int_as_float(__builtin_amdgcn_readlane(__float_as_int(wL), i));
        float v = feat[s * 32 + lane] * w;
        atomicAdd(agg + d * 32 + lane, v);
    }
}

// ---------------------------------------------------------------------------
// h = relu(agg + b) in place
// ---------------------------------------------------------------------------
__global__ void relu_bias_kernel(float* __restrict__ h, const float* __restrict__ b, int n32) {
    int idx = blockIdx.x * blockDim.x + threadIdx.x;
    if (idx < n32) {
        float v = h[idx] + b[idx & 31];
        h[idx] = v > 0.0f ? v : 0.0f;
    }
}

// ---------------------------------------------------------------------------
// Fused output GEMM: mu = A@Wmu + bmu, lv = A@Wlv + blv  (A = agg2 [N,32]).
// Shares A fragments across both 16x16x(K=32) products.
// d_out layout: [mu (N*16) | logvar (N*16)].
// ---------------------------------------------------------------------------
__global__ void gemm_out_kernel(const float* __restrict__ A,
                                const float* __restrict__ Wmu, const float* __restrict__ bmu,
                                const float* __restrict__ Wlv, const float* __restrict__ blv,
                                float* __restrict__ out, int nRows) {
    int waveId = (blockIdx.x * blockDim.x + threadIdx.x) >> 5;
    int lane   = threadIdx.x & 31;
    int m0     = waveId * 16;
    if (m0 >= nRows) return;

    int row  = m0 + (lane & 15);
    int grp  = lane >> 4;
    int koff = grp * 2;
    int col  = lane & 15;

    v8f accM = {};
    v8f accL = {};
#pragma unroll
    for (int k = 0; k < 8; ++k) {
        int kb = 4 * k + koff;
        v2f a = *reinterpret_cast<const v2f*>(A + row * 32 + kb);
        v2f bm, bl;
        bm.x = Wmu[kb * 16 + col];  bm.y = Wmu[(kb + 1) * 16 + col];
        bl.x = Wlv[kb * 16 + col];  bl.y = Wlv[(kb + 1) * 16 + col];
        accM = __builtin_amdgcn_wmma_f32_16x16x4_f32(false, a, false, bm, (short)0, accM, false, false);
        accL = __builtin_amdgcn_wmma_f32_16x16x4_f32(false, a, false, bl, (short)0, accL, false, false);
    }

    int rowbase = m0 + grp * 8;
    float vbm = bmu[col];
    float vbl = blv[col];
    float* outLv = out + (size_t)nRows * 16;
#pragma unroll
    for (int r = 0; r < 8; ++r) {
        out[(rowbase + r) * 16 + col]   = accM[r] + vbm;
        outLv[(rowbase + r) * 16 + col] = accL[r] + vbl;
    }
}

// ---------------------------------------------------------------------------
// Launch sequence
// ---------------------------------------------------------------------------
extern "C" void kernel_launch(void* const* d_in, const int* in_sizes, int n_in,
                              void* d_out, int out_size, void* d_ws, size_t ws_size,
                              hipStream_t stream) {
    const float* x   = (const float*)d_in[0];
    const int*   ei  = (const int*)  d_in[1];
    const float* W1  = (const float*)d_in[2];
    const float* b1  = (const float*)d_in[3];
    const float* Wmu = (const float*)d_in[4];
    const float* bmu = (const float*)d_in[5];
    const float* Wlv = (const float*)d_in[6];
    const float* blv = (const float*)d_in[7];

    const int N = in_sizes[0] / 32;   // 100000
    const int E = in_sizes[1] / 2;    // 1600000
    const int* src = ei;              // edge_index[0]
    const int* dst = ei + E;          // edge_index[1]

    // Workspace: dinv[N] | bufA[N*32] | bufB[N*32]  (~26 MB)
    float* dinv = (float*)d_ws;
    float* bufA = dinv + N;
    float* bufB = bufA + (size_t)N * 32;
    float* out  = (float*)d_out;

    const int TB = 256;
    const int n32 = N * 32;
    const int gemmThreads = ((N + 15) / 16) * 32;            // one wave per 16-row tile
    const int scatterWaves = (E + 31) / 32;                  // one wave per 32-edge batch
    const int scatterBlocks = (scatterWaves * 32 + TB - 1) / TB;

    // 1) degrees (init with self-loop) -> dinv = deg^-1/2
    deg_init_kernel<<<(N + TB - 1) / TB, TB, 0, stream>>>(dinv, N);
    deg_count_kernel<<<(E + TB - 1) / TB, TB, 0, stream>>>(dst, dinv, E);
    deg_to_dinv_kernel<<<(N + TB - 1) / TB, TB, 0, stream>>>(dinv, N);

    // 2) layer 1: xw = x @ W1 (WMMA), aggregate, h = relu(agg + b1)
    gemm_xw1_kernel<<<(gemmThreads + TB - 1) / TB, TB, 0, stream>>>(x, W1, bufA, N);
    self_init_kernel<<<(n32 + TB - 1) / TB, TB, 0, stream>>>(bufA, dinv, bufB, n32);
    scatter_agg_kernel<<<scatterBlocks, TB, 0, stream>>>(src, dst, dinv, bufA, bufB, E);
    relu_bias_kernel<<<(n32 + TB - 1) / TB, TB, 0, stream>>>(bufB, b1, n32);

    // 3) layer 2: aggregate h once (linearity => GEMM after aggregation),
    //    then mu / logvar via fused WMMA GEMM with bias.
    self_init_kernel<<<(n32 + TB - 1) / TB, TB, 0, stream>>>(bufB, dinv, bufA, n32);
    scatter_agg_kernel<<<scatterBlocks, TB, 0, stream>>>(src, dst, dinv, bufB, bufA, E);
    gemm_out_kernel<<<(gemmThreads + TB - 1) / TB, TB, 0, stream>>>(bufA, Wmu, bmu, Wlv, blv, out, N);
}